// Model_44332652429635
// MI455X (gfx1250) — compile-verified
//
#include <hip/hip_runtime.h>
#include <math.h>

// ---- problem constants (match setup_inputs) ----
#define LDIM   16
#define CDIM   224      // 128+64+32 heights
#define WDIM   221      // 127+63+31 widths
#define LOG2PI_F   1.83787706640934534f
#define LOGGTOT_F  9.90348755253612804f   // log(20000)

typedef float v2f __attribute__((ext_vector_type(2)));
typedef float v8f __attribute__((ext_vector_type(8)));

// ---------------- reduction helpers ----------------
__device__ __forceinline__ float warpSumF(float v) {
  #pragma unroll
  for (int off = 16; off > 0; off >>= 1) v += __shfl_xor(v, off, 32);
  return v;
}
__device__ __forceinline__ int warpSumI(int v) {
  #pragma unroll
  for (int off = 16; off > 0; off >>= 1) v += __shfl_xor(v, off, 32);
  return v;
}

// =====================================================================
// Kernel A: per-gene spline precompute (width softmax -> loc/w), gather uh,
// plus  -sum(normal_logpdf(rho_weight[genes_oi]))  contribution.
// =====================================================================
__global__ void precompute_genes(const int* __restrict__ genes_oi,
                                 const float* __restrict__ uh,
                                 const float* __restrict__ uw,
                                 const float* __restrict__ rho_weight,
                                 float* __restrict__ ws_loc,
                                 float* __restrict__ ws_w,
                                 float* __restrict__ ws_uh,
                                 float* __restrict__ out,
                                 int n_genes) {
  int gl = blockIdx.x * blockDim.x + threadIdx.x;
  float acc = 0.f;
  if (gl < n_genes) {
    int g0 = genes_oi[gl];
    // gather unnormalized heights for this gene-of-interest
    for (int c = 0; c < CDIM; ++c)
      ws_uh[(size_t)gl * CDIM + c] = uh[(size_t)g0 * CDIM + c];
    // rho_weight[genes_oi] prior term:  elbo += 0.5*v^2 + 0.5*log(2pi)
    for (int l = 0; l < LDIM; ++l) {
      float v = rho_weight[(size_t)g0 * LDIM + l];
      acc += 0.5f * v * v + 0.5f * LOG2PI_F;
    }
    const int ns[3] = {128, 64, 32};
    int oh = 0, ow = 0;
    for (int s = 0; s < 3; ++s) {
      int n = ns[s], nw = n - 1;
      const float* u = uw + (size_t)g0 * WDIM + ow;
      float m = -3.0e38f;
      for (int j = 0; j < nw; ++j) m = fmaxf(m, u[j]);
      float ssum = 0.f;
      for (int j = 0; j < nw; ++j) ssum += expf(u[j] - m);
      float inv = 1.f / ssum;
      float cum = 0.f;
      float* locp = ws_loc + (size_t)gl * CDIM + oh;
      float* wp   = ws_w   + (size_t)gl * WDIM + ow;
      locp[0] = 0.f;
      for (int j = 0; j < nw; ++j) {
        float wj = expf(u[j] - m) * inv;
        wp[j] = wj;
        cum += wj;
        locp[j + 1] = (j == nw - 1) ? 1.0f : cum;   // bl.at[:,-1].set(1.0)
      }
      oh += n; ow += nw;
    }
  }
  // block reduce + atomic
  __shared__ float smem[32];
  float v = warpSumF(acc);
  int lane = threadIdx.x & 31, wid = threadIdx.x >> 5;
  if (lane == 0) smem[wid] = v;
  __syncthreads();
  if (threadIdx.x == 0) {
    float s = 0.f; int nwv = (blockDim.x + 31) >> 5;
    for (int i = 0; i < nwv; ++i) s += smem[i];
    atomicAdd(out, s);
  }
}

// =====================================================================
// Kernel B: per-batch-row logsumexp of (log rho_bias + clustering@rho_w^T)
// WMMA f32 16x16x4, K=16 chained as 4 steps. One wave (32 thr) per block;
// block handles 16 batch rows, streams all genes in 16-wide tiles.
// =====================================================================
__global__ void rho_lse_kernel(const float* __restrict__ clustering,
                               const float* __restrict__ rho_weight,
                               const float* __restrict__ rho_bias,
                               float* __restrict__ ws_lse,
                               int Gt) {
  int lane = threadIdx.x;
  int rowbase = blockIdx.x * 16;
  int m    = lane & 15;          // matrix row (A) / col (B) within tile
  int koff = (lane >> 4) * 2;    // half-wave K offset per ISA 16x4 layout

  // A tiles (clustering rows) are loop-invariant: preload for kk=0..3
  v2f a[4];
  #pragma unroll
  for (int kk = 0; kk < 4; ++kk) {
    int kb = kk * 4 + koff;
    a[kk].x = clustering[(size_t)(rowbase + m) * LDIM + kb];
    a[kk].y = clustering[(size_t)(rowbase + m) * LDIM + kb + 1];
  }

  float M[8], S[8];
  #pragma unroll
  for (int r = 0; r < 8; ++r) { M[r] = -3.0e38f; S[r] = 0.f; }

  for (int gt = 0; gt < Gt; gt += 16) {
    v8f c = {};
    #pragma unroll
    for (int kk = 0; kk < 4; ++kk) {
      int kb = kk * 4 + koff;
      v2f b;
      b.x = rho_weight[(size_t)(gt + m) * LDIM + kb];
      b.y = rho_weight[(size_t)(gt + m) * LDIM + kb + 1];
      c = __builtin_amdgcn_wmma_f32_16x16x4_f32(
            /*neg_a=*/false, a[kk], /*neg_b=*/false, b,
            /*c_mod=*/(short)0, c, /*reuse_a=*/false, /*reuse_b=*/false);
    }
    float lb = logf(rho_bias[gt + m]);
    // C/D layout: component r, lanes 0-15 -> row rowbase+r, lanes16-31 -> +8
    #pragma unroll
    for (int r = 0; r < 8; ++r) {
      float z  = c[r] + lb;
      float Mn = fmaxf(M[r], z);
      S[r] = S[r] * expf(M[r] - Mn) + expf(z - Mn);
      M[r] = Mn;
    }
  }
  // merge (M,S) across the 16 lanes of each half-wave
  #pragma unroll
  for (int mask = 1; mask < 16; mask <<= 1) {
    #pragma unroll
    for (int r = 0; r < 8; ++r) {
      float Mo = __shfl_xor(M[r], mask, 32);
      float So = __shfl_xor(S[r], mask, 32);
      float Mn = fmaxf(M[r], Mo);
      S[r] = S[r] * expf(M[r] - Mn) + So * expf(Mo - Mn);
      M[r] = Mn;
    }
  }
  if ((lane & 15) == 0) {
    int half = lane >> 4;
    #pragma unroll
    for (int r = 0; r < 8; ++r)
      ws_lse[rowbase + r + 8 * half] = M[r] + logf(S[r]);
  }
}

// =====================================================================
// Kernel D:  -sum(normal_logpdf(logit_weight[genes_oi]))  contribution.
// =====================================================================
__global__ void lw_npdf_kernel(const float* __restrict__ logit_weight,
                               const int* __restrict__ genes_oi,
                               float* __restrict__ out, int n_genes) {
  const int per = LDIM * CDIM;                 // 3584
  long total = (long)n_genes * per;
  float acc = 0.f;
  for (long i = (long)blockIdx.x * blockDim.x + threadIdx.x; i < total;
       i += (long)gridDim.x * blockDim.x) {
    int gl  = (int)(i / per);
    int rem = (int)(i - (long)gl * per);
    float v = logit_weight[(size_t)genes_oi[gl] * per + rem];
    acc += 0.5f * v * v + 0.5f * LOG2PI_F;
  }
  __shared__ float smem[32];
  float v = warpSumF(acc);
  int lane = threadIdx.x & 31, wid = threadIdx.x >> 5;
  if (lane == 0) smem[wid] = v;
  __syncthreads();
  if (threadIdx.x == 0) {
    float s = 0.f; int nwv = (blockDim.x + 31) >> 5;
    for (int i = 0; i < nwv; ++i) s += smem[i];
    atomicAdd(out, s);
  }
}

// =====================================================================
// Kernel C: main per-cut kernel. One wave per cut (grid-stride over waves).
//  - rho logit for this cut (16-dot + precomputed lse)
//  - delta row = clustering[b] . logit_weight[g]  (224 cols, L2-resident)
//  - 3-stage quadratic spline: exp heights, wave scan for CDF, uniform bin
// =====================================================================
__global__ void cuts_kernel(const float* __restrict__ clustering,
                            const float* __restrict__ coordinates,
                            const int* __restrict__ genes_oi,
                            const int* __restrict__ local_gene_ix,
                            const int* __restrict__ local_cellxgene_ix,
                            const int* __restrict__ localcellxgene_ix,
                            const float* __restrict__ logit_weight,
                            const float* __restrict__ rho_weight,
                            const float* __restrict__ rho_bias,
                            const float* __restrict__ ws_loc,
                            const float* __restrict__ ws_w,
                            const float* __restrict__ ws_uh,
                            const float* __restrict__ ws_lse,
                            float* __restrict__ out,
                            int N, int n_genes, int Gt) {
  int lane = threadIdx.x & 31;
  int wloc = threadIdx.x >> 5;
  int wave = blockIdx.x * (blockDim.x >> 5) + wloc;
  int nwaves = gridDim.x * (blockDim.x >> 5);
  float acc = 0.f;

  for (int n = wave; n < N; n += nwaves) {
    // ---- overall (rho) likelihood ----
    int idx2 = localcellxgene_ix[n];
    int b2 = idx2 / Gt, g2 = idx2 - b2 * Gt;
    float p = (lane < LDIM)
      ? clustering[(size_t)b2 * LDIM + lane] * rho_weight[(size_t)g2 * LDIM + lane]
      : 0.f;
    p = warpSumF(p);
    float lrho = logf(rho_bias[g2]) + p - ws_lse[b2] + LOGGTOT_F;

    // ---- mixture delta row (224 cols; c = lane + 32*j) ----
    int idx1 = local_cellxgene_ix[n];
    int b1 = idx1 / n_genes, gl1 = idx1 - b1 * n_genes;
    int g1 = genes_oi[gl1];
    float delta[7];
    #pragma unroll
    for (int j = 0; j < 7; ++j) delta[j] = 0.f;
    const float* lwrow = logit_weight + (size_t)g1 * LDIM * CDIM;
    const float* crow  = clustering + (size_t)b1 * LDIM;
    #pragma unroll 4
    for (int l = 0; l < LDIM; ++l) {
      float cv = crow[l];
      const float* lr = lwrow + l * CDIM;
      #pragma unroll
      for (int j = 0; j < 7; ++j)
        delta[j] = fmaf(cv, lr[lane + 32 * j], delta[j]);
    }

    // ---- spline stack ----
    int gls = local_gene_ix[n];
    const float* uhrow  = ws_uh  + (size_t)gls * CDIM;
    const float* locrow = ws_loc + (size_t)gls * CDIM;
    const float* wrow   = ws_w   + (size_t)gls * WDIM;

    float x = coordinates[n];
    float lad = 0.f;
    int oh = 0, ow = 0, cb = 0;
    const int ns[3] = {128, 64, 32};
    #pragma unroll
    for (int s = 0; s < 3; ++s) {
      const int nn = ns[s];
      const int nch = nn >> 5;
      float h[4], seg[4], cum[4], wv[4], lv[4];
      #pragma unroll
      for (int jc = 0; jc < nch; ++jc) {
        int idx = lane + 32 * jc;
        h[jc]  = expf(uhrow[oh + idx] + delta[cb + jc]);
        lv[jc] = locrow[oh + idx];
        wv[jc] = (idx < nn - 1) ? wrow[ow + idx] : 0.f;
      }
      // trapezoid segments: seg[i] = 0.5*(h[i]+h[i+1])*w[i]
      #pragma unroll
      for (int jc = 0; jc < nch; ++jc) {
        float hn     = __shfl_down(h[jc], 1, 32);
        float hnext0 = (jc + 1 < nch) ? __shfl(h[jc + 1], 0, 32) : 0.f;
        if (lane == 31) hn = hnext0;
        int idx = lane + 32 * jc;
        seg[jc] = (idx < nn - 1) ? 0.5f * (h[jc] + hn) * wv[jc] : 0.f;
      }
      // inclusive scan across chunks (area = grand total; seg[nn-1]==0)
      float carry = 0.f;
      #pragma unroll
      for (int jc = 0; jc < nch; ++jc) {
        float v = seg[jc];
        #pragma unroll
        for (int off = 1; off < 32; off <<= 1) {
          float t = __shfl_up(v, off, 32);
          if (lane >= off) v += t;
        }
        cum[jc] = v + carry;
        carry = __shfl(cum[jc], 31, 32);
      }
      float area = carry;
      // bin index (wave-uniform: x is per-cut scalar)
      int cnt = 0;
      #pragma unroll
      for (int jc = 0; jc < nch; ++jc) cnt += (x >= lv[jc]) ? 1 : 0;
      cnt = warpSumI(cnt);
      int bin = cnt - 1;
      bin = min(max(bin, 0), nn - 2);
      int bl = bin & 31, bc = bin >> 5;
      int bp = bin + 1;
      int bl2 = bp & 31, bc2 = bp >> 5;
      float hsel  = (bc  == 0) ? h[0] : (bc  == 1) ? h[1] : (bc  == 2) ? h[2] : h[3];
      float hsel2 = (bc2 == 0) ? h[0] : (bc2 == 1) ? h[1] : (bc2 == 2) ? h[2] : h[3];
      float csel  = (bc == 0) ? (cum[0] - seg[0]) : (bc == 1) ? (cum[1] - seg[1])
                  : (bc == 2) ? (cum[2] - seg[2]) : (cum[3] - seg[3]);
      float lsel  = (bc == 0) ? lv[0] : (bc == 1) ? lv[1] : (bc == 2) ? lv[2] : lv[3];
      float wsel  = (bc == 0) ? wv[0] : (bc == 1) ? wv[1] : (bc == 2) ? wv[2] : wv[3];
      float h_l_un = __shfl(hsel,  bl,  32);
      float h_r_un = __shfl(hsel2, bl2, 32);
      float cdf_un = __shfl(csel,  bl,  32);
      float loc_l  = __shfl(lsel,  bl,  32);
      float w_b    = __shfl(wsel,  bl,  32);
      float inv = 1.f / area;
      float h_l = h_l_un * inv, h_r = h_r_un * inv, cdf_l = cdf_un * inv;
      float alpha = (x - loc_l) / w_b;
      x = 0.5f * (h_r - h_l) * w_b * alpha * alpha + h_l * w_b * alpha + cdf_l;
      lad += logf(h_l + (h_r - h_l) * alpha);
      oh += nn; ow += nn - 1; cb += nch;
    }
    acc += -(lad + lrho);   // elbo = -sum(lik_mixture + lik_overall)
  }

  // per-wave acc is uniform; reduce one value per wave then atomic per block
  __shared__ float smem[32];
  if (lane == 0) smem[wloc] = acc;
  __syncthreads();
  if (threadIdx.x == 0) {
    float s = 0.f; int nwv = blockDim.x >> 5;
    for (int i = 0; i < nwv; ++i) s += smem[i];
    atomicAdd(out, s);
  }
}

// =====================================================================
extern "C" void kernel_launch(void* const* d_in, const int* in_sizes, int n_in,
                              void* d_out, int out_size, void* d_ws, size_t ws_size,
                              hipStream_t stream) {
  const float* clustering        = (const float*)d_in[0];
  const float* coordinates       = (const float*)d_in[1];
  const int*   genes_oi          = (const int*)  d_in[2];
  const int*   local_gene_ix     = (const int*)  d_in[3];
  const int*   local_cellxgene_ix= (const int*)  d_in[4];
  const int*   localcellxgene_ix = (const int*)  d_in[5];
  const float* logit_weight      = (const float*)d_in[6];
  const float* rho_weight        = (const float*)d_in[7];
  const float* rho_bias          = (const float*)d_in[8];
  const float* uh                = (const float*)d_in[9];
  const float* uw                = (const float*)d_in[10];

  int B       = in_sizes[0] / LDIM;   // 512
  int N       = in_sizes[1];          // 200000
  int n_genes = in_sizes[2];          // 500
  int Gt      = in_sizes[8];          // 20000

  float* out = (float*)d_out;

  // workspace layout (floats)
  float* ws_loc = (float*)d_ws;                      // n_genes * 224
  float* ws_w   = ws_loc + (size_t)n_genes * CDIM;   // n_genes * 221
  float* ws_uh  = ws_w   + (size_t)n_genes * WDIM;   // n_genes * 224
  float* ws_lse = ws_uh  + (size_t)n_genes * CDIM;   // B

  hipMemsetAsync(out, 0, sizeof(float), stream);

  // A: per-gene spline precompute + rho_weight prior term
  {
    int threads = 128;
    int blocks = (n_genes + threads - 1) / threads;
    precompute_genes<<<blocks, threads, 0, stream>>>(
        genes_oi, uh, uw, rho_weight, ws_loc, ws_w, ws_uh, out, n_genes);
  }
  // B: WMMA rho-delta GEMM + per-row logsumexp
  {
    rho_lse_kernel<<<B / 16, 32, 0, stream>>>(
        clustering, rho_weight, rho_bias, ws_lse, Gt);
  }
  // D: logit_weight[genes_oi] prior term
  {
    lw_npdf_kernel<<<256, 256, 0, stream>>>(logit_weight, genes_oi, out, n_genes);
  }
  // C: main per-cut likelihood
  {
    cuts_kernel<<<640, 256, 0, stream>>>(
        clustering, coordinates, genes_oi, local_gene_ix,
        local_cellxgene_ix, localcellxgene_ix, logit_weight,
        rho_weight, rho_bias, ws_loc, ws_w, ws_uh, ws_lse,
        out, N, n_genes, Gt);
  }
}